// HashEmbedder_42494406427409
// MI455X (gfx1250) — compile-verified
//
#include <hip/hip_runtime.h>
#include <math.h>

// ---------------------------------------------------------------------------
// Instant-NGP style multiresolution hash encoding, forward.
//   out[p, 2*lvl + f] = sum_{c=0..7} cw[p,lvl,c] * table[lvl][hash(corner_c)][f]
//   mask[p]           = all(|coords| <= 1)
// Mapping: one thread per (point, level) -> 8.4M threads, wave32-friendly,
// output stores fully coalesced (2*tid indexing).
// Tables (64MB) are kept L2-resident (RT loads); coords/out/mask streamed
// with non-temporal hints so they don't evict the tables from the 192MB L2.
// ---------------------------------------------------------------------------

#define NPTS   524288u
#define NLVL   16
#define TSIZE  (1u << 19)
#define TMASK  (TSIZE - 1u)
#define PRIME1 2654435761u
#define PRIME2 805459861u

typedef float v2f __attribute__((ext_vector_type(2)));

struct GridArgs { float g[NLVL]; };

// Tiny init kernel: put the 16 per-level grid sizes (computed on host with the
// exact float64 exp/log/pow/floor sequence numpy uses) into d_ws so the main
// kernel can index them per-lane.
__global__ void hash_grid_init_kernel(GridArgs a, float* __restrict__ ws) {
  if (blockIdx.x == 0 && threadIdx.x == 0) {
#pragma unroll
    for (int i = 0; i < NLVL; ++i) ws[i] = a.g[i];
  }
}

__global__ __launch_bounds__(256) void hash_embed_kernel(
    const float* __restrict__ coords,   // [NPTS,3]
    const float* __restrict__ tables,   // [NLVL, TSIZE, 2]
    const float* __restrict__ gridtab,  // [NLVL] grid cell sizes (f32)
    float* __restrict__ out,            // [NPTS, 32]
    float* __restrict__ mask)           // [NPTS]
{
  const unsigned tid = blockIdx.x * 256u + threadIdx.x; // = p*16 + lvl
  const unsigned lvl = tid & (NLVL - 1u);
  const unsigned p   = tid >> 4;
  if (p >= NPTS) return;

  // Streamed (read-once) coords: non-temporal so they don't pollute L2.
  const float cx = __builtin_nontemporal_load(coords + 3u * p + 0u);
  const float cy = __builtin_nontemporal_load(coords + 3u * p + 1u);
  const float cz = __builtin_nontemporal_load(coords + 3u * p + 2u);

  if (lvl == 0u) {
    const bool keep = (cx >= -1.0f) && (cx <= 1.0f) &&
                      (cy >= -1.0f) && (cy <= 1.0f) &&
                      (cz >= -1.0f) && (cz <= 1.0f);
    __builtin_nontemporal_store(keep ? 1.0f : 0.0f, mask + p);
  }

  const float grid = gridtab[lvl];

  // Clamp (matches jnp.clip), bottom-left corner via exact f32 div + floor.
  const float ccx = fminf(fmaxf(cx, -1.0f), 1.0f);
  const float ccy = fminf(fmaxf(cy, -1.0f), 1.0f);
  const float ccz = fminf(fmaxf(cz, -1.0f), 1.0f);

  const float bx = floorf((ccx + 1.0f) / grid);
  const float by = floorf((ccy + 1.0f) / grid);
  const float bz = floorf((ccz + 1.0f) / grid);

  // Reference uses the ORIGINAL (unclamped) coords for the weights.
  const float wx = (cx - (bx * grid - 1.0f)) / grid;
  const float wy = (cy - (by * grid - 1.0f)) / grid;
  const float wz = (cz - (bz * grid - 1.0f)) / grid;

  // Spatial hash: int64 math in the reference, but (x ^ y*P1 ^ z*P2) & 0x7FFFF
  // depends only on the low 32 bits -> exact with uint32 arithmetic.
  const unsigned ix = (unsigned)(int)bx;
  const unsigned iy = (unsigned)(int)by;
  const unsigned iz = (unsigned)(int)bz;

  const unsigned hx0 = ix;               // * PRIMES[0] == 1
  const unsigned hx1 = ix + 1u;
  const unsigned hy0 = iy * PRIME1;
  const unsigned hy1 = hy0 + PRIME1;
  const unsigned hz0 = iz * PRIME2;
  const unsigned hz1 = hz0 + PRIME2;

  // Corner order: c = i*4 + j*2 + k  (i->x, j->y, k->z), matching OFFSETS.
  const v2f* __restrict__ tbl = (const v2f*)tables + (size_t)lvl * (size_t)TSIZE;
  const v2f e0 = tbl[(hx0 ^ hy0 ^ hz0) & TMASK];
  const v2f e1 = tbl[(hx0 ^ hy0 ^ hz1) & TMASK];
  const v2f e2 = tbl[(hx0 ^ hy1 ^ hz0) & TMASK];
  const v2f e3 = tbl[(hx0 ^ hy1 ^ hz1) & TMASK];
  const v2f e4 = tbl[(hx1 ^ hy0 ^ hz0) & TMASK];
  const v2f e5 = tbl[(hx1 ^ hy0 ^ hz1) & TMASK];
  const v2f e6 = tbl[(hx1 ^ hy1 ^ hz0) & TMASK];
  const v2f e7 = tbl[(hx1 ^ hy1 ^ hz1) & TMASK];

  // Trilinear corner weights, product order ((x*y)*z) as in the reference.
  const float u0 = 1.0f - wx, u1 = wx;
  const float v0 = 1.0f - wy, v1 = wy;
  const float s0 = 1.0f - wz, s1 = wz;
  const float w00 = u0 * v0, w01 = u0 * v1, w10 = u1 * v0, w11 = u1 * v1;

  v2f acc = e0 * (w00 * s0);
  acc += e1 * (w00 * s1);
  acc += e2 * (w01 * s0);
  acc += e3 * (w01 * s1);
  acc += e4 * (w10 * s0);
  acc += e5 * (w10 * s1);
  acc += e6 * (w11 * s0);
  acc += e7 * (w11 * s1);

  // out[p*32 + 2*lvl] == out[2*tid] -> wave stores 256 contiguous bytes.
  __builtin_nontemporal_store(acc, (v2f*)out + (size_t)tid);
}

extern "C" void kernel_launch(void* const* d_in, const int* in_sizes, int n_in,
                              void* d_out, int out_size, void* d_ws, size_t ws_size,
                              hipStream_t stream) {
  (void)in_sizes; (void)n_in; (void)out_size; (void)ws_size;

  // Mirror numpy's float64 math exactly:
  //   b = exp(log(512/16)/15); res_i = floor(16 * b**i); grid_i = f32(2.0/res_i)
  // Done on the host with the same libm calls numpy uses, since floor() here
  // sits 1 ulp from an integer at i = 3,6,9,12,15.
  GridArgs ga;
  const double b = exp(log(512.0 / 16.0) / 15.0);
  for (int i = 0; i < NLVL; ++i) {
    const double res = floor(16.0 * pow(b, (double)i));
    ga.g[i] = (float)(2.0 / res);
  }

  const float* coords = (const float*)d_in[0];
  const float* tables = (const float*)d_in[1];
  float* out  = (float*)d_out;
  float* mask = out + (size_t)NPTS * 32u;  // tuple output: [N,32] then [N]
  float* ws   = (float*)d_ws;

  hipLaunchKernelGGL(hash_grid_init_kernel, dim3(1), dim3(32), 0, stream, ga, ws);

  const unsigned total  = NPTS * (unsigned)NLVL;      // 8,388,608
  const unsigned blocks = total / 256u;               // 32,768
  hipLaunchKernelGGL(hash_embed_kernel, dim3(blocks), dim3(256), 0, stream,
                     coords, tables, ws, out, mask);
}